// PermutationRandom_29858612642211
// MI455X (gfx1250) — compile-verified
//
#include <hip/hip_runtime.h>
#include <stdint.h>

#define B_DIM 256
#define L_DIM 2048
#define D_DIM 128
#define RPW   8      // rows per wave in the gather (MLP depth)
#define WAVES 8      // waves per block (256 threads, wave32)

// ---------------- threefry2x32 (JAX-style uniform keys) ----------------
__device__ __forceinline__ uint32_t rotl32(uint32_t x, uint32_t r) {
  return (x << r) | (x >> (32u - r));
}

__device__ __forceinline__ void threefry2x32(uint32_t k0, uint32_t k1,
                                             uint32_t c0, uint32_t c1,
                                             uint32_t& o0, uint32_t& o1) {
  const uint32_t ks2 = 0x1BD11BDAu ^ k0 ^ k1;
  uint32_t ks[3] = {k0, k1, ks2};
  const uint32_t rotA[4] = {13u, 15u, 26u, 6u};
  const uint32_t rotB[4] = {17u, 29u, 16u, 24u};
  uint32_t x0 = c0 + k0;
  uint32_t x1 = c1 + k1;
#pragma unroll
  for (int g = 0; g < 5; ++g) {
    const uint32_t* r = (g & 1) ? rotB : rotA;
#pragma unroll
    for (int i = 0; i < 4; ++i) {
      x0 += x1;
      x1 = rotl32(x1, r[i]);
      x1 ^= x0;
    }
    x0 += ks[(g + 1) % 3];
    x1 += ks[(g + 2) % 3] + (uint32_t)(g + 1);
  }
  o0 = x0;
  o1 = x1;
}

__device__ __forceinline__ float bits_to_uniform(uint32_t bits) {
  const uint32_t u = (bits >> 9) | 0x3F800000u;
  return __uint_as_float(u) - 1.0f;
}

// uniform value at flat index into a (B,L) array, key = (0, 42)
__device__ __forceinline__ float jax_uniform_at(uint32_t flat) {
  const uint32_t n = (uint32_t)B_DIM * (uint32_t)L_DIM;
  const uint32_t half = n / 2u;
  uint32_t c0, c1;
  int sel;
  if (flat < half) { c0 = flat; c1 = flat + half; sel = 0; }
  else             { c0 = flat - half; c1 = flat; sel = 1; }
  uint32_t o0, o1;
  threefry2x32(0u, 42u, c0, c1, o0, o1);
  return bits_to_uniform(sel ? o1 : o0);
}

// ---------------- kernel 1: per-row bitonic sort of (key, idx) ----------------
__global__ void __launch_bounds__(256)
sort_perm_kernel(const int* __restrict__ mask, float* __restrict__ perm_out) {
  __shared__ float skey[L_DIM];
  __shared__ int   sidx[L_DIM];
  const int b = blockIdx.x;

  for (int i = threadIdx.x; i < L_DIM; i += 256) {
    float k;
    if (mask[b * L_DIM + i] == 1)
      k = jax_uniform_at((uint32_t)(b * L_DIM + i));    // real: random key in [0,1)
    else
      k = 2.0f + (float)i * (1.0f / (float)L_DIM);      // pad: > any real, ascending
    skey[i] = k;
    sidx[i] = i;
  }
  __syncthreads();

  // ascending bitonic sort over 2048 elements resident in LDS
  for (int ssize = 2; ssize <= L_DIM; ssize <<= 1) {
    for (int j = ssize >> 1; j > 0; j >>= 1) {
      for (int i = threadIdx.x; i < L_DIM; i += 256) {
        const int ixj = i ^ j;
        if (ixj > i) {
          const bool asc = ((i & ssize) == 0);
          const float a = skey[i];
          const float c = skey[ixj];
          if ((a > c) == asc) {
            skey[i] = c; skey[ixj] = a;
            const int t = sidx[i]; sidx[i] = sidx[ixj]; sidx[ixj] = t;
          }
        }
      }
      __syncthreads();
    }
  }

  for (int i = threadIdx.x; i < L_DIM; i += 256)
    perm_out[b * L_DIM + i] = (float)sidx[i];
}

// ---------------- kernel 2: pipelined async-LDS gather ----------------
// Each wave32 copies RPW=8 consecutive 512B output rows. It issues 8
// non-temporal async global->LDS loads back-to-back (ASYNCcnt=8, 4KB in
// flight per wave), waits once, then fires 8 async LDS->global stores and
// exits (S_ENDPGM implicitly waits for all counters).
__global__ void __launch_bounds__(256)
gather_async_kernel(const float* __restrict__ x,
                    const float* __restrict__ permf,
                    float* __restrict__ out) {
  __shared__ __align__(16) unsigned char stage[WAVES * RPW * 512];  // 32 KB

  const uint32_t wave = threadIdx.x >> 5;
  const uint32_t lane = threadIdx.x & 31u;
  const uint32_t row_base = (blockIdx.x * WAVES + wave) * RPW;

  // LDS byte offset of this lane's slot in row-chunk 0 (low 32b of generic
  // LDS address == LDS offset per ISA aperture rules)
  const uint32_t lds0 =
      (uint32_t)(unsigned long long)(uintptr_t)(&stage[(wave * RPW) * 512u + lane * 16u]);

  // perm entries are wave-uniform (consecutive rows) -> scalarizable loads
  int p[RPW];
#pragma unroll
  for (int k = 0; k < RPW; ++k)
    p[k] = (int)permf[row_base + k];

  // stage all RPW source rows into LDS (streaming, non-temporal)
#pragma unroll
  for (int k = 0; k < RPW; ++k) {
    const uint32_t row = row_base + (uint32_t)k;
    const uint32_t b = row >> 11;  // L = 2048
    const unsigned long long src =
        (unsigned long long)(uintptr_t)(x + ((size_t)b * L_DIM + (size_t)p[k]) * D_DIM) +
        (unsigned long long)(lane * 16u);
    const uint32_t ldsk = lds0 + (uint32_t)(k * 512);
    asm volatile("global_load_async_to_lds_b128 %0, %1, off th:TH_LOAD_NT"
                 :: "v"(ldsk), "v"(src)
                 : "memory");
  }

  // all loads complete in order among themselves; one wait drains them all
  asm volatile("s_wait_asynccnt 0" ::: "memory");

  // drain LDS -> global (non-temporal streaming stores)
#pragma unroll
  for (int k = 0; k < RPW; ++k) {
    const uint32_t row = row_base + (uint32_t)k;
    const unsigned long long dst =
        (unsigned long long)(uintptr_t)(out + (size_t)row * D_DIM) +
        (unsigned long long)(lane * 16u);
    const uint32_t ldsk = lds0 + (uint32_t)(k * 512);
    asm volatile("global_store_async_from_lds_b128 %0, %1, off th:TH_STORE_NT"
                 :: "v"(dst), "v"(ldsk)
                 : "memory");
  }
  // no trailing wait: S_ENDPGM performs an implicit wait-idle
}

// ---------------- launcher ----------------
extern "C" void kernel_launch(void* const* d_in, const int* in_sizes, int n_in,
                              void* d_out, int out_size, void* d_ws, size_t ws_size,
                              hipStream_t stream) {
  (void)in_sizes; (void)n_in; (void)out_size; (void)d_ws; (void)ws_size;
  const float* x    = (const float*)d_in[0];
  const int*   mask = (const int*)d_in[1];
  float* out      = (float*)d_out;
  float* perm_out = out + (size_t)B_DIM * (size_t)L_DIM * (size_t)D_DIM;  // tuple tail

  sort_perm_kernel<<<B_DIM, 256, 0, stream>>>(mask, perm_out);

  const int rows = B_DIM * L_DIM;                    // 524288 rows
  const int blocks = rows / (WAVES * RPW);           // 8192 blocks
  gather_async_kernel<<<blocks, 256, 0, stream>>>(x, perm_out, out);
}